// HeteroGNN_61392262529298
// MI455X (gfx1250) — compile-verified
//
#include <hip/hip_runtime.h>
#include <hip/hip_bf16.h>

typedef __attribute__((ext_vector_type(2))) float v2f;
typedef __attribute__((ext_vector_type(8))) float v8f;

#define N_MOVIE    100000
#define N_DIRECTOR 40000
#define N_ACTOR    150000
#define D_IN  128
#define D_HID 128
#define D_OUT 64
#define E_MD 200000
#define E_DM 200000
#define E_MA 600000
#define E_AM 600000

// ---------------- zero ----------------
__global__ void zero_kernel(float* __restrict__ p, long long n) {
    long long i = (long long)blockIdx.x * blockDim.x + threadIdx.x;
    long long stride = (long long)gridDim.x * blockDim.x;
    for (; i < n; i += stride) p[i] = 0.0f;
}

// ---------------- weight transpose: in[4][K][dout] -> out[4][dout][K] ----------------
__global__ void transpose_w_kernel(const float* __restrict__ in, float* __restrict__ out,
                                   int K, int dout) {
    int total = 4 * K * dout;
    int i = blockIdx.x * blockDim.x + threadIdx.x;
    int stride = gridDim.x * blockDim.x;
    for (; i < total; i += stride) {
        int r = i / (K * dout);
        int t = i - r * K * dout;
        int k = t / dout;
        int j = t - k * dout;
        out[r * K * dout + j * K + k] = in[i];
    }
}

// ---------------- degree count ----------------
__global__ void count_kernel(const int* __restrict__ dst, int E, float* __restrict__ cnt) {
    int i = blockIdx.x * blockDim.x + threadIdx.x;
    int stride = gridDim.x * blockDim.x;
    for (; i < E; i += stride) atomicAdd(&cnt[dst[i]], 1.0f);
}

// ---------------- edge scatter-add (128-wide rows), one wave per edge ----------------
__global__ void scatter_add_kernel(const float* __restrict__ x,
                                   const int* __restrict__ src,
                                   const int* __restrict__ dst,
                                   int E, float* __restrict__ agg) {
    int wave  = (blockIdx.x * blockDim.x + threadIdx.x) >> 5;
    int lane  = threadIdx.x & 31;
    int nwave = (gridDim.x * blockDim.x) >> 5;
    for (int e = wave; e < E; e += nwave) {
        int s = src[e];
        int d = dst[e];
        const float4 v = *(const float4*)(x + (long long)s * 128 + lane * 4);
        float* p = agg + (long long)d * 128 + lane * 4;
        atomicAdd(p + 0, v.x);
        atomicAdd(p + 1, v.y);
        atomicAdd(p + 2, v.z);
        atomicAdd(p + 3, v.w);
    }
}

// ---------------- mean: agg[row][:] /= max(cnt[row],1) ----------------
__global__ void scale_kernel(float* __restrict__ agg, const float* __restrict__ cnt, int n) {
    long long total = (long long)n * 128;
    long long i = (long long)blockIdx.x * blockDim.x + threadIdx.x;
    long long stride = (long long)gridDim.x * blockDim.x;
    for (; i < total; i += stride) {
        int row = (int)(i >> 7);
        agg[i] *= (1.0f / fmaxf(cnt[row], 1.0f));
    }
}

// ---------------- fused 2-input WMMA GEMM (transposed weights) ----------------
// out[n,dout] (+)= A1[n,128] @ W1 + A2[n,128] @ W2 + bias[dout]
// W1t/W2t are column-major transposed: Wt[col][k], k contiguous -> B fragment is one b64 load.
// One wave computes a 16x16 tile: 32 x v_wmma_f32_16x16x4_f32 per input, fully unrolled.
// blockDim.x = 32 * (dout/16); blockIdx.x = row tile. All waves fully active (EXEC all ones).
__global__ void gemm2_wmma_kernel(const float* __restrict__ A1, const float* __restrict__ W1t,
                                  const float* __restrict__ A2, const float* __restrict__ W2t,
                                  const float* __restrict__ bias, float* __restrict__ out,
                                  int n, int dout, int accumulate, int relu) {
    const int lane    = threadIdx.x & 31;
    const int half2   = (lane >> 4) << 1;   // 0 or 2 : lane's K-pair offset within each K=4 step
    const int l16     = lane & 15;
    const int colTile = threadIdx.x >> 5;
    const int row0    = blockIdx.x * 16;
    const int col     = colTile * 16 + l16;

    v8f acc = {};

    // ---- A1 @ W1 ----
    {
        const float* ap = A1 + (long long)(row0 + l16) * 128 + half2;
        const float* bp = W1t + (long long)col * 128 + half2;
        #pragma unroll
        for (int k = 0; k < 128; k += 4) {
            v2f a = *(const v2f*)(ap + k);
            v2f b = *(const v2f*)(bp + k);
            acc = __builtin_amdgcn_wmma_f32_16x16x4_f32(false, a, false, b,
                                                        (short)0, acc, false, false);
        }
    }
    // ---- + A2 @ W2 ----
    {
        const float* ap = A2 + (long long)(row0 + l16) * 128 + half2;
        const float* bp = W2t + (long long)col * 128 + half2;
        #pragma unroll
        for (int k = 0; k < 128; k += 4) {
            v2f a = *(const v2f*)(ap + k);
            v2f b = *(const v2f*)(bp + k);
            acc = __builtin_amdgcn_wmma_f32_16x16x4_f32(false, a, false, b,
                                                        (short)0, acc, false, false);
        }
    }

    const float bval = bias[col];
    const int halfRow = (lane >> 4) * 8;
    #pragma unroll
    for (int i = 0; i < 8; ++i) {
        int row = row0 + i + halfRow;   // C/D layout: vgpr i -> M=i (lanes 0-15), M=8+i (lanes 16-31)
        long long idx = (long long)row * dout + col;
        float v = acc[i] + bval;
        if (accumulate) v += out[idx];
        if (relu) v = fmaxf(v, 0.0f);
        out[idx] = v;
    }
}

static void launch_gemm2(const float* A1, const float* W1t, const float* A2, const float* W2t,
                         const float* bias, float* out, int n, int dout,
                         int accumulate, int relu, hipStream_t stream) {
    dim3 block(32 * (dout / 16));
    dim3 grid(n / 16);
    gemm2_wmma_kernel<<<grid, block, 0, stream>>>(A1, W1t, A2, W2t, bias, out, n, dout, accumulate, relu);
}

extern "C" void kernel_launch(void* const* d_in, const int* in_sizes, int n_in,
                              void* d_out, int out_size, void* d_ws, size_t ws_size,
                              hipStream_t stream) {
    const float* x_m = (const float*)d_in[0];
    const float* x_d = (const float*)d_in[1];
    const float* x_a = (const float*)d_in[2];
    const float* W1l = (const float*)d_in[3];   // [4,128,128]
    const float* W1r = (const float*)d_in[4];   // [4,128,128]
    const float* b1  = (const float*)d_in[5];   // [4,128]
    const float* W2l = (const float*)d_in[6];   // [4,128,64]
    const float* W2r = (const float*)d_in[7];   // [4,128,64]
    const float* b2  = (const float*)d_in[8];   // [4,64]
    const int* ei_md = (const int*)d_in[9];     // [2, E] : src row then dst row
    const int* ei_dm = (const int*)d_in[10];
    const int* ei_ma = (const int*)d_in[11];
    const int* ei_am = (const int*)d_in[12];

    // ---- workspace layout (floats) ----
    float* ws = (float*)d_ws;
    long long off = 0;
    float* agg_d  = ws + off; off += (long long)N_DIRECTOR * 128;  // m->d aggregate
    float* agg_dm = ws + off; off += (long long)N_MOVIE    * 128;  // d->m aggregate
    float* agg_am = ws + off; off += (long long)N_MOVIE    * 128;  // a->m aggregate
    float* agg_a  = ws + off; off += (long long)N_ACTOR    * 128;  // m->a aggregate
    const long long AGG_TOTAL = off;
    float* cnt_d  = ws + off; off += N_DIRECTOR;
    float* cnt_dm = ws + off; off += N_MOVIE;
    float* cnt_am = ws + off; off += N_MOVIE;
    float* cnt_a  = ws + off; off += N_ACTOR;
    const long long AGGCNT_TOTAL = off;
    float* h_m    = ws + off; off += (long long)N_MOVIE    * 128;
    float* h_d    = ws + off; off += (long long)N_DIRECTOR * 128;
    float* h_a    = ws + off; off += (long long)N_ACTOR    * 128;
    // transposed weights [4][dout][128]
    float* W1lT   = ws + off; off += 4LL * 128 * 128;
    float* W1rT   = ws + off; off += 4LL * 128 * 128;
    float* W2lT   = ws + off; off += 4LL * 128 * 64;
    float* W2rT   = ws + off; off += 4LL * 128 * 64;

    float* o_m = (float*)d_out;                                    // [100000,64]
    float* o_d = o_m + (long long)N_MOVIE * D_OUT;                 // [40000,64]
    float* o_a = o_d + (long long)N_DIRECTOR * D_OUT;              // [150000,64]

    const int ZB = 2048, ZT = 256;

    // ---- pre-transpose weights (tiny) ----
    transpose_w_kernel<<<256, 256, 0, stream>>>(W1l, W1lT, 128, 128);
    transpose_w_kernel<<<256, 256, 0, stream>>>(W1r, W1rT, 128, 128);
    transpose_w_kernel<<<128, 256, 0, stream>>>(W2l, W2lT, 128, 64);
    transpose_w_kernel<<<128, 256, 0, stream>>>(W2r, W2rT, 128, 64);

    // ===== layer 1 =====
    zero_kernel<<<ZB, ZT, 0, stream>>>(ws, AGGCNT_TOTAL);

    count_kernel<<<512, 256, 0, stream>>>(ei_md + E_MD, E_MD, cnt_d);
    count_kernel<<<512, 256, 0, stream>>>(ei_dm + E_DM, E_DM, cnt_dm);
    count_kernel<<<512, 256, 0, stream>>>(ei_am + E_AM, E_AM, cnt_am);
    count_kernel<<<512, 256, 0, stream>>>(ei_ma + E_MA, E_MA, cnt_a);

    // one wave per edge: blocks of 256 = 8 waves
    scatter_add_kernel<<<(E_MD + 7) / 8, 256, 0, stream>>>(x_m, ei_md, ei_md + E_MD, E_MD, agg_d);
    scatter_add_kernel<<<(E_DM + 7) / 8, 256, 0, stream>>>(x_d, ei_dm, ei_dm + E_DM, E_DM, agg_dm);
    scatter_add_kernel<<<(E_AM + 7) / 8, 256, 0, stream>>>(x_a, ei_am, ei_am + E_AM, E_AM, agg_am);
    scatter_add_kernel<<<(E_MA + 7) / 8, 256, 0, stream>>>(x_m, ei_ma, ei_ma + E_MA, E_MA, agg_a);

    scale_kernel<<<ZB, ZT, 0, stream>>>(agg_d,  cnt_d,  N_DIRECTOR);
    scale_kernel<<<ZB, ZT, 0, stream>>>(agg_dm, cnt_dm, N_MOVIE);
    scale_kernel<<<ZB, ZT, 0, stream>>>(agg_am, cnt_am, N_MOVIE);
    scale_kernel<<<ZB, ZT, 0, stream>>>(agg_a,  cnt_a,  N_ACTOR);

    const long long W1SZ = 128 * 128, B1SZ = 128;
    // h_d = relu(sage rel0)
    launch_gemm2(agg_d,  W1lT + 0 * W1SZ, x_d, W1rT + 0 * W1SZ, b1 + 0 * B1SZ, h_d, N_DIRECTOR, D_HID, 0, 1, stream);
    // h_a = relu(sage rel2)
    launch_gemm2(agg_a,  W1lT + 2 * W1SZ, x_a, W1rT + 2 * W1SZ, b1 + 2 * B1SZ, h_a, N_ACTOR,    D_HID, 0, 1, stream);
    // h_m = relu(sage rel1 + sage rel3)
    launch_gemm2(agg_dm, W1lT + 1 * W1SZ, x_m, W1rT + 1 * W1SZ, b1 + 1 * B1SZ, h_m, N_MOVIE,    D_HID, 0, 0, stream);
    launch_gemm2(agg_am, W1lT + 3 * W1SZ, x_m, W1rT + 3 * W1SZ, b1 + 3 * B1SZ, h_m, N_MOVIE,    D_HID, 1, 1, stream);

    // ===== layer 2 (counts reused) =====
    zero_kernel<<<ZB, ZT, 0, stream>>>(ws, AGG_TOTAL);

    scatter_add_kernel<<<(E_MD + 7) / 8, 256, 0, stream>>>(h_m, ei_md, ei_md + E_MD, E_MD, agg_d);
    scatter_add_kernel<<<(E_DM + 7) / 8, 256, 0, stream>>>(h_d, ei_dm, ei_dm + E_DM, E_DM, agg_dm);
    scatter_add_kernel<<<(E_AM + 7) / 8, 256, 0, stream>>>(h_a, ei_am, ei_am + E_AM, E_AM, agg_am);
    scatter_add_kernel<<<(E_MA + 7) / 8, 256, 0, stream>>>(h_m, ei_ma, ei_ma + E_MA, E_MA, agg_a);

    scale_kernel<<<ZB, ZT, 0, stream>>>(agg_d,  cnt_d,  N_DIRECTOR);
    scale_kernel<<<ZB, ZT, 0, stream>>>(agg_dm, cnt_dm, N_MOVIE);
    scale_kernel<<<ZB, ZT, 0, stream>>>(agg_am, cnt_am, N_MOVIE);
    scale_kernel<<<ZB, ZT, 0, stream>>>(agg_a,  cnt_a,  N_ACTOR);

    const long long W2SZ = 128 * 64, B2SZ = 64;
    launch_gemm2(agg_d,  W2lT + 0 * W2SZ, h_d, W2rT + 0 * W2SZ, b2 + 0 * B2SZ, o_d, N_DIRECTOR, D_OUT, 0, 0, stream);
    launch_gemm2(agg_a,  W2lT + 2 * W2SZ, h_a, W2rT + 2 * W2SZ, b2 + 2 * B2SZ, o_a, N_ACTOR,    D_OUT, 0, 0, stream);
    launch_gemm2(agg_dm, W2lT + 1 * W2SZ, h_m, W2rT + 1 * W2SZ, b2 + 1 * B2SZ, o_m, N_MOVIE,    D_OUT, 0, 0, stream);
    launch_gemm2(agg_am, W2lT + 3 * W2SZ, h_m, W2rT + 3 * W2SZ, b2 + 3 * B2SZ, o_m, N_MOVIE,    D_OUT, 1, 0, stream);
}